// COCQCNN_layer_44229573214630
// MI455X (gfx1250) — compile-verified
//
#include <hip/hip_runtime.h>
#include <hip/hip_bf16.h>
#include <math.h>

// ============================================================================
// 5-qubit quantum patch simulator on MI455X (gfx1250).
//
// State per patch: 32 complex amps -> real 64-vector [Re(0..31); Im(0..31)].
// Wire w -> amplitude-index bit: wire0=16, wire1=8, wire2=4, wire3=2, wire4=1.
//
// Shared gates per layer (CU1..CU4, CPhase) are composed on-device into a
// real 64x64 matrix G_l = [[Re U, -Im U],[Im U, Re U]] (prologue kernel).
// Main kernel applies G_l to 32 patches/wave via V_WMMA_F32_16X16X4_F32:
//   D[64x32] = G[64x64] x S[64x32]   (4 M-tiles, 2 N-tiles, 16 K-steps)
// Per-patch RX gates + H + measurement run per-lane in VALU registers,
// with an LDS round-trip as the layout shim between per-lane and the
// wave-striped WMMA layout (ISA 7.12.2).
// ============================================================================

typedef __attribute__((ext_vector_type(2))) float v2f;
typedef __attribute__((ext_vector_type(8))) float v8f;

#define RS2 0.70710678118654752f

// ----------------------------------------------------------------------------
// Prologue: build G[3][64][64] from thetas[12], phis[3].
// Thread t: layer l = t/32, basis column j = t%32. Push e_j through
// CU(1,2) CU(2,3) CU(3,4) CU(4,1) then CPhase(0,1); emit real 64x64 columns.
// X_theta = [[0, -i e^{i t/2}],[-i e^{-i t/2}, 0]] applied when control=1.
// ----------------------------------------------------------------------------
__global__ __launch_bounds__(96) void qc_build_G(
    const float* __restrict__ thetas, const float* __restrict__ phis,
    float* __restrict__ G) {
  int t = threadIdx.x;
  if (t >= 96) return;
  int l = t >> 5;
  int j = t & 31;

  float re[32], im[32];
#pragma unroll
  for (int i = 0; i < 32; ++i) { re[i] = 0.0f; im[i] = 0.0f; }
  re[j] = 1.0f;

  const int cb[4] = {8, 4, 2, 1};  // control bits: wires 1,2,3,4
  const int tb[4] = {4, 2, 1, 8};  // target  bits: wires 2,3,4,1
#pragma unroll
  for (int w = 0; w < 4; ++w) {
    float th = thetas[4 * l + w];
    float ec = cosf(0.5f * th), es = sinf(0.5f * th);
    // -i*e^{+i th/2} = (es, -ec) ; -i*e^{-i th/2} = (-es, -ec)
    int bc = cb[w], bt = tb[w];
#pragma unroll
    for (int i = 0; i < 32; ++i) {
      if ((i & bc) && !(i & bt)) {
        int i1 = i | bt;
        float ur = re[i], ui = im[i], vr = re[i1], vi = im[i1];
        re[i]  =  es * vr + ec * vi;   // (es - i ec) * v
        im[i]  =  es * vi - ec * vr;
        re[i1] = -es * ur + ec * ui;   // (-es - i ec) * u
        im[i1] = -es * ui - ec * ur;
      }
    }
  }
  // CPhase(phi) on wires 0,1 -> bits 16 & 8
  float pr = cosf(phis[l]), pi = sinf(phis[l]);
#pragma unroll
  for (int i = 0; i < 32; ++i) {
    if ((i & 16) && (i & 8)) {
      float r = re[i] * pr - im[i] * pi;
      float m = re[i] * pi + im[i] * pr;
      re[i] = r; im[i] = m;
    }
  }
  // Real 64x64 embedding: col j = [Re; Im], col j+32 = [-Im; Re]
  float* Gl = G + l * 64 * 64;
#pragma unroll
  for (int r = 0; r < 32; ++r) {
    Gl[r * 64 + j]             =  re[r];
    Gl[r * 64 + j + 32]        = -im[r];
    Gl[(r + 32) * 64 + j]      =  im[r];
    Gl[(r + 32) * 64 + j + 32] =  re[r];
  }
}

// ----------------------------------------------------------------------------
// Main kernel: 128 threads = 4 waves, 1 patch per lane (32 patches/wave).
// ----------------------------------------------------------------------------
#define GP 65  // LDS pitch for staged G (bank-conflict-free A-frag reads)
#define SP 33  // LDS pitch for per-wave state tile

__global__ __launch_bounds__(128) void qc_main(
    const float* __restrict__ x, const float* __restrict__ G,
    float* __restrict__ out, int P) {
  __shared__ float Gs[64 * GP];        // current layer's G, padded
  __shared__ float Sb[4][64 * SP];     // per-wave 64x32 state tile, padded

  const int tid  = threadIdx.x;
  const int lane = tid & 31;
  const int wv   = tid >> 5;
  const int m    = lane & 15;          // column-within-N-tile / row-within-half
  const int hi   = lane >> 4;          // lane half (WMMA layout)
  float* Sw = &Sb[wv][0];

  const int p  = blockIdx.x * 128 + tid;   // patch id == output flat index
  const int b  = p >> 14;                  // 128*128 patches per image
  const int h2 = (p >> 7) & 127;
  const int w2 = p & 127;

  // 12 per-patch angles: ang[c*4 + dy*2 + dx] = x[b][c][2h2+dy][2w2+dx]
  float ang[12];
#pragma unroll
  for (int c = 0; c < 3; ++c) {
#pragma unroll
    for (int dy = 0; dy < 2; ++dy) {
      const float* row = x + (((b * 3 + c) * 256 + (2 * h2 + dy)) * 256 + 2 * w2);
      ang[c * 4 + dy * 2 + 0] = row[0];
      ang[c * 4 + dy * 2 + 1] = row[1];
    }
  }

  // State after H on wire 0: amp(0)=amp(16)=1/sqrt(2)
  float st[64];
#pragma unroll
  for (int i = 0; i < 64; ++i) st[i] = 0.0f;
  st[0] = RS2;
  st[16] = RS2;

#pragma unroll
  for (int l = 0; l < 3; ++l) {
    // ---- per-patch RX on wires 1..4 (bits 8,4,2,1), in registers ----
#pragma unroll
    for (int wq = 0; wq < 4; ++wq) {
      float h = 0.5f * ang[4 * l + wq];
      float c_ = cosf(h), s_ = sinf(h);
      const int bt = 8 >> wq;
#pragma unroll
      for (int i = 0; i < 32; ++i) {
        if (!(i & bt)) {
          int jj = i | bt;
          float ur = st[i], ui = st[i + 32], vr = st[jj], vi = st[jj + 32];
          st[i]       = c_ * ur + s_ * vi;
          st[i + 32]  = c_ * ui - s_ * vr;
          st[jj]      = c_ * vr + s_ * ui;
          st[jj + 32] = c_ * vi - s_ * ur;
        }
      }
    }

    __syncthreads();  // prior layer's Gs/Sw reads done before restaging

    // ---- stage G_l into LDS (coalesced, padded) ----
    for (int idx = tid; idx < 4096; idx += 128) {
      int r = idx >> 6, cc = idx & 63;
      Gs[r * GP + cc] = G[(l * 64 + r) * 64 + cc];
    }
    // ---- scatter per-lane state into wave tile: S[row][lane] ----
#pragma unroll
    for (int r = 0; r < 64; ++r) Sw[r * SP + lane] = st[r];

    __syncthreads();

    // ---- D[64x32] = G_l x S via V_WMMA_F32_16X16X4_F32 ----
    // A 16x4 frag (ISA 7.12.2): v0 = G[m][k0+2*hi], v1 = G[m][k0+1+2*hi]
    // B 4x16 frag (contiguous-K-block pattern): v0 = S[k0+2*hi][n], v1 = S[k0+1+2*hi][n]
    v8f acc[4][2] = {};
    for (int ks = 0; ks < 16; ++ks) {
      const int kA = ks * 4 + 2 * hi;
      v2f Bf0, Bf1;
      Bf0[0] = Sw[(kA + 0) * SP + m];
      Bf0[1] = Sw[(kA + 1) * SP + m];
      Bf1[0] = Sw[(kA + 0) * SP + 16 + m];
      Bf1[1] = Sw[(kA + 1) * SP + 16 + m];
#pragma unroll
      for (int mt = 0; mt < 4; ++mt) {
        v2f Af;
        Af[0] = Gs[(mt * 16 + m) * GP + kA];
        Af[1] = Gs[(mt * 16 + m) * GP + kA + 1];
        acc[mt][0] = __builtin_amdgcn_wmma_f32_16x16x4_f32(
            false, Af, false, Bf0, (short)0, acc[mt][0], false, false);
        acc[mt][1] = __builtin_amdgcn_wmma_f32_16x16x4_f32(
            false, Af, false, Bf1, (short)0, acc[mt][1], false, false);
      }
    }

    // ---- write D back (C/D layout: VGPR r -> rows r / r+8 per lane half) ----
#pragma unroll
    for (int mt = 0; mt < 4; ++mt)
#pragma unroll
      for (int nt = 0; nt < 2; ++nt)
#pragma unroll
        for (int r = 0; r < 8; ++r)
          Sw[(mt * 16 + hi * 8 + r) * SP + nt * 16 + m] = acc[mt][nt][r];

    // same-wave LDS ops are in-order; gather back to per-lane state
#pragma unroll
    for (int r = 0; r < 64; ++r) st[r] = Sw[r * SP + lane];
  }

  // ---- final H on wire 0 (bit 16) ----
#pragma unroll
  for (int i = 0; i < 16; ++i) {
    float ur = st[i],      ui = st[i + 32];
    float vr = st[i + 16], vi = st[i + 48];
    st[i]      = (ur + vr) * RS2;
    st[i + 32] = (ui + vi) * RS2;
    st[i + 16] = (ur - vr) * RS2;
    st[i + 48] = (ui - vi) * RS2;
  }

  // ---- <Z0> = sum_i sign(bit16) * |amp_i|^2 ----
  float ev = 0.0f;
#pragma unroll
  for (int i = 0; i < 32; ++i) {
    float p2 = st[i] * st[i] + st[i + 32] * st[i + 32];
    ev += (i & 16) ? -p2 : p2;
  }
  if (p < P) out[p] = ev;
}

// ----------------------------------------------------------------------------
extern "C" void kernel_launch(void* const* d_in, const int* in_sizes, int n_in,
                              void* d_out, int out_size, void* d_ws, size_t ws_size,
                              hipStream_t stream) {
  (void)in_sizes; (void)n_in; (void)ws_size;
  const float* x      = (const float*)d_in[0];  // [16,3,256,256]
  const float* thetas = (const float*)d_in[1];  // [12]
  const float* phis   = (const float*)d_in[2];  // [3]
  float* out = (float*)d_out;                   // [16,1,128,128]
  float* G   = (float*)d_ws;                    // [3][64][64] f32 = 48 KB

  hipLaunchKernelGGL(qc_build_G, dim3(1), dim3(96), 0, stream, thetas, phis, G);

  const int P = out_size;            // 262144, multiple of 128
  const int blocks = (P + 127) / 128;
  hipLaunchKernelGGL(qc_main, dim3(blocks), dim3(128), 0, stream, x, G, out, P);
}